// SAGEModule_88364657148502
// MI455X (gfx1250) — compile-verified
//
#include <hip/hip_runtime.h>
#include <hip/hip_bf16.h>

// SAGEConv (mean aggregation), fp32, MI455X / gfx1250.
//   inputs: x[N,128] f32, edge_index[2,E] i64, W_l[128,128] f32, W_r[128,128] f32, b[128] f32
//   out_i = relu( W_l @ mean_{j in N(i)} x_j + W_r @ x_i + b )
// Workspace layout (d_ws): agg[N*128] f32 | cnt[N] f32   (~51.6 MB)
//
// 3 kernels: zero-ws -> scatter-add (edges, f32 atomics, degree fused) ->
//            fused [mean-divide + dual WMMA-f32 GEMM + bias + relu].

#define DIM 128

typedef __attribute__((ext_vector_type(2))) float v2f;
typedef __attribute__((ext_vector_type(8))) float v8f;

// ---------------------------------------------------------------- zero ws ---
__global__ __launch_bounds__(256)
void sage_zero_kernel(float4* __restrict__ ws, long long n_vec4) {
    long long i = (long long)blockIdx.x * blockDim.x + threadIdx.x;
    long long stride = (long long)gridDim.x * blockDim.x;
    float4 z = make_float4(0.f, 0.f, 0.f, 0.f);
    for (; i < n_vec4; i += stride) ws[i] = z;
}

// ------------------------------------------------------------ scatter-add ---
// 32 lanes per edge; each lane float4-gathers from x[src] and does 4
// global_atomic_add_f32 into agg[dst]; lane 0 of each edge also bumps cnt.
// x (51MB) and agg (51MB) are both L2-resident (192MB L2), so the 80M atomics
// run at L2 atomic throughput — this is the dominant phase of the workload.
__global__ __launch_bounds__(256)
void sage_scatter_kernel(const float* __restrict__ x, const long long* __restrict__ ei,
                         float* __restrict__ agg, float* __restrict__ cnt,
                         int n_edges) {
    long long tid = (long long)blockIdx.x * blockDim.x + threadIdx.x;
    if (tid >= (long long)n_edges * 32) return;
    int e = (int)(tid >> 5);
    int g = (int)(tid & 31);                    // 4-float group within the row
    long long s = ei[e];                        // src (gather)
    long long d = ei[(long long)n_edges + e];   // dst (scatter)
    const float4 v = *(const float4*)&x[(size_t)s * DIM + g * 4];
    float* p = &agg[(size_t)d * DIM + g * 4];
    atomicAdd(p + 0, v.x);
    atomicAdd(p + 1, v.y);
    atomicAdd(p + 2, v.z);
    atomicAdd(p + 3, v.w);
    if (g == 0) atomicAdd(&cnt[d], 1.0f);       // fused degree count
}

// --------------------------------- fused mean + dual WMMA f32 GEMM + relu ---
// out = relu( (agg/max(cnt,1)) @ W_l^T + x @ W_r^T + b )
// Block = 256 threads = 8 wave32; block owns 16 rows, wave w owns cols [16w,16w+16).
// V_WMMA_F32_16X16X4_F32 fragment layouts (ISA 7.12.2):
//   A (16x4):  lane l -> row m=l%16, holds K = {k0+2*(l/16), +1} in {v.x,v.y}
//   B (4x16):  lane l -> col n=l%16, holds K = {k0+2*(l/16), +1} in {v.x,v.y}
//   C/D (16x16): VGPR v, lane l -> row v+8*(l/16), col l%16
// B must be W^T (out = A @ W^T), so B[k][n] = W[n*DIM + k]: a contiguous
// float2 straight out of W's row-major storage.
__global__ __launch_bounds__(256)
void sage_gemm_kernel(const float* __restrict__ x, const float* __restrict__ agg,
                      const float* __restrict__ cnt,
                      const float* __restrict__ Wl, const float* __restrict__ Wr,
                      const float* __restrict__ bias, float* __restrict__ out,
                      int n_nodes) {
    __shared__ float lds_m[16 * DIM];
    __shared__ float lds_x[16 * DIM];

    const int rowBase = blockIdx.x * 16;
    const int tid = threadIdx.x;

    // Stage 16 rows into LDS (2 x 8KB); mean division fused into the load so
    // agg never needs a separate read-modify-write pass over 51MB.
    for (int i = tid; i < 16 * DIM / 4; i += 256) {
        int r = i >> 5;                         // (i*4)/DIM
        int c = (i << 2) & (DIM - 1);
        int rr = rowBase + r;
        if (rr >= n_nodes) rr = n_nodes - 1;    // clamp (N%16==0 in practice)
        float deg   = cnt[rr];
        float scale = 1.0f / fmaxf(deg, 1.0f);
        float4 a = *(const float4*)&agg[(size_t)rr * DIM + c];
        a.x *= scale; a.y *= scale; a.z *= scale; a.w *= scale;
        *(float4*)&lds_m[i * 4] = a;
        *(float4*)&lds_x[i * 4] = *(const float4*)&x[(size_t)rr * DIM + c];
    }
    __syncthreads();

    const int wave = tid >> 5;                  // 0..7  (wave32)
    const int lane = tid & 31;
    const int half = lane >> 4;                 // 0 or 1
    const int m    = lane & 15;                 // A row within tile
    const int n    = (lane & 15) + wave * 16;   // output column 0..127

    // Warm caches for this wave's 16 weight rows (global_prefetch_b8).
    __builtin_prefetch(&Wl[(size_t)n * DIM], 0, 3);
    __builtin_prefetch(&Wr[(size_t)n * DIM], 0, 3);

    v8f acc = {};
    #pragma unroll 4
    for (int k0 = 0; k0 < DIM; k0 += 4) {
        const int kk = k0 + 2 * half;
        v2f a_m = *(const v2f*)&lds_m[m * DIM + kk];
        v2f b_l = *(const v2f*)&Wl[(size_t)n * DIM + kk];
        acc = __builtin_amdgcn_wmma_f32_16x16x4_f32(false, a_m, false, b_l,
                                                    (short)0, acc, false, false);
        v2f a_x = *(const v2f*)&lds_x[m * DIM + kk];
        v2f b_r = *(const v2f*)&Wr[(size_t)n * DIM + kk];
        acc = __builtin_amdgcn_wmma_f32_16x16x4_f32(false, a_x, false, b_r,
                                                    (short)0, acc, false, false);
    }

    const float bv = bias[n];
    if (rowBase + 16 <= n_nodes) {
        // Fast path: uniform, no exec-mask churn; stores coalesce across the
        // 16 lanes of each half-wave (64B per row segment).
        #pragma unroll
        for (int v = 0; v < 8; ++v) {
            float val = acc[v] + bv;
            out[(size_t)(rowBase + v + 8 * half) * DIM + n] = fmaxf(val, 0.0f);
        }
    } else {
        #pragma unroll
        for (int v = 0; v < 8; ++v) {
            int row = rowBase + v + 8 * half;
            if (row < n_nodes) {
                float val = acc[v] + bv;
                out[(size_t)row * DIM + n] = fmaxf(val, 0.0f);
            }
        }
    }
}

// ------------------------------------------------------------------ launch ---
extern "C" void kernel_launch(void* const* d_in, const int* in_sizes, int n_in,
                              void* d_out, int out_size, void* d_ws, size_t ws_size,
                              hipStream_t stream) {
    const float*     x  = (const float*)d_in[0];
    const long long* ei = (const long long*)d_in[1];   // int64 [2, E]
    const float*     Wl = (const float*)d_in[2];
    const float*     Wr = (const float*)d_in[3];
    const float*     b  = (const float*)d_in[4];
    float*           out = (float*)d_out;

    const int n_nodes = in_sizes[0] / DIM;
    const int n_edges = in_sizes[1] / 2;

    float* agg = (float*)d_ws;                               // [N*DIM]
    float* cnt = agg + (size_t)n_nodes * DIM;                // [N]

    // 1) zero agg+cnt (N*(DIM+1) floats; N%4==0 so it's float4-exact)
    long long n_vec4 = ((long long)n_nodes * (DIM + 1)) / 4;
    {
        int blocks = (int)((n_vec4 + 255) / 256);
        if (blocks > 65535) blocks = 65535;                  // grid-stride
        sage_zero_kernel<<<blocks, 256, 0, stream>>>((float4*)d_ws, n_vec4);
    }
    // 2) scatter-add messages + fused degree count
    {
        long long work = (long long)n_edges * 32;
        sage_scatter_kernel<<<(int)((work + 255) / 256), 256, 0, stream>>>(
            x, ei, agg, cnt, n_edges);
    }
    // 3) fused mean + dual-GEMM (V_WMMA_F32_16X16X4_F32) + bias + relu
    sage_gemm_kernel<<<(n_nodes + 15) / 16, 256, 0, stream>>>(
        x, agg, cnt, Wl, Wr, b, out, n_nodes);
}